// GATEncoder_6932077216357
// MI455X (gfx1250) — compile-verified
//
#include <hip/hip_runtime.h>
#include <hip/hip_bf16.h>
#include <math.h>

typedef __attribute__((ext_vector_type(2))) float v2f;
typedef __attribute__((ext_vector_type(8))) float v8f;

#define NEG_SLOPE 0.2f
#define BN_EPS 1e-5f

// ---------------------------------------------------------------------------
// f32 WMMA GEMM: Hout[M,Nout] = X[M,K] @ W[K,Nout]
// One wave computes one 16x16 output tile via V_WMMA_F32_16X16X4_F32.
// A-frag layout (16x4 f32): lanes 0-15 hold rows, half=lane>>4 selects K pair.
// B-frag layout (4x16 f32): lanes hold columns, half selects K pair.
// C/D layout (16x16 f32, 8 VGPRs): vgpr r -> row (half*8 + r), col = lane&15.
// ---------------------------------------------------------------------------
__global__ void gemm_wmma_f32(const float* __restrict__ X,
                              const float* __restrict__ W,
                              float* __restrict__ Hout,
                              int M, int K, int Nout) {
  int wid  = (int)((blockIdx.x * blockDim.x + threadIdx.x) >> 5);
  int lane = threadIdx.x & 31;
  int ntiles = Nout >> 4;
  int mt = wid / ntiles;
  int nt = wid - mt * ntiles;
  if (mt * 16 >= M) return;
  int half = lane >> 4;
  int l    = lane & 15;

  const float* arow = X + (size_t)(mt * 16 + l) * K + half * 2;
  const float* bcol = W + nt * 16 + l;

  v8f acc = {};
  for (int kb = 0; kb < K; kb += 4) {
    v2f a, b;
    a.x = arow[kb];
    a.y = arow[kb + 1];
    int k0 = kb + half * 2;
    b.x = bcol[(size_t)k0 * Nout];
    b.y = bcol[(size_t)(k0 + 1) * Nout];
    acc = __builtin_amdgcn_wmma_f32_16x16x4_f32(false, a, false, b, (short)0,
                                                acc, false, false);
  }

  int row0 = mt * 16 + half * 8;
  int col  = nt * 16 + l;
  float* outp = Hout + (size_t)row0 * Nout + col;
#pragma unroll
  for (int r = 0; r < 8; r++) outp[(size_t)r * Nout] = acc[r];
}

// ---------------------------------------------------------------------------
__global__ void fill_f32(float* __restrict__ p, float v, int n) {
  int t = blockIdx.x * blockDim.x + threadIdx.x;
  if (t < n) p[t] = v;
}

// a_src[n,h] = <h[n,h,:], att_src[h,:]> ; a_dst likewise
__global__ void attn_coef(const float* __restrict__ h,
                          const float* __restrict__ att_s,
                          const float* __restrict__ att_d,
                          float* __restrict__ as_out,
                          float* __restrict__ ad_out,
                          int Nn, int H, int C) {
  int t = blockIdx.x * blockDim.x + threadIdx.x;
  if (t >= Nn * H) return;
  int n  = t / H;
  int hh = t - n * H;
  const float* hp = h + (size_t)n * H * C + (size_t)hh * C;
  const float* sp = att_s + (size_t)hh * C;
  const float* dp = att_d + (size_t)hh * C;
  float s = 0.f, dv = 0.f;
  for (int c = 0; c < C; c++) {
    float v = hp[c];
    s  += v * sp[c];
    dv += v * dp[c];
  }
  as_out[t] = s;
  ad_out[t] = dv;
}

// float atomic max via monotone int/uint mapping (works with mixed signs)
__device__ __forceinline__ void atomicMaxFloat(float* addr, float val) {
  if (val >= 0.f) {
    atomicMax((int*)addr, __float_as_int(val));
  } else {
    atomicMin((unsigned int*)addr, __float_as_uint(val));
  }
}

__device__ __forceinline__ void edge_sd(int e, int E,
                                        const long long* __restrict__ src,
                                        const long long* __restrict__ dst,
                                        int& s, int& d) {
  if (e < E) {
    s = (int)src[e];
    d = (int)dst[e];
  } else {
    s = d = e - E;  // appended self-loop
  }
}

// pass 1: alpha = leaky_relu(a_src[s]+a_dst[d]); segment max over dst
__global__ void edge_pass1(const long long* __restrict__ src,
                           const long long* __restrict__ dst,
                           int E, int Etot,
                           const float* __restrict__ as_,
                           const float* __restrict__ ad_,
                           float* __restrict__ alpha,
                           float* __restrict__ amax, int H) {
  int t = blockIdx.x * blockDim.x + threadIdx.x;
  if (t >= Etot * H) return;
  int e  = t / H;
  int hh = t - e * H;
  int s, d;
  edge_sd(e, E, src, dst, s, d);
  float v = as_[s * H + hh] + ad_[d * H + hh];
  v = v > 0.f ? v : NEG_SLOPE * v;
  alpha[t] = v;
  atomicMaxFloat(&amax[d * H + hh], v);
}

// pass 2: alpha = exp(alpha - amax[d]); segment sum -> denom
__global__ void edge_pass2(const long long* __restrict__ src,
                           const long long* __restrict__ dst,
                           int E, int Etot,
                           float* __restrict__ alpha,
                           const float* __restrict__ amax,
                           float* __restrict__ denom, int H) {
  int t = blockIdx.x * blockDim.x + threadIdx.x;
  if (t >= Etot * H) return;
  int e  = t / H;
  int hh = t - e * H;
  int s, d;
  edge_sd(e, E, src, dst, s, d);
  float w = expf(alpha[t] - amax[d * H + hh]);
  alpha[t] = w;
  atomicAdd(&denom[d * H + hh], w);
}

// pass 3: agg[d,:] += h[s,:] * alpha_norm   (block per edge, thread per chan)
__global__ void edge_scatter(const long long* __restrict__ src,
                             const long long* __restrict__ dst,
                             int E,
                             const float* __restrict__ h,
                             const float* __restrict__ alpha,
                             const float* __restrict__ denom,
                             float* __restrict__ agg, int H, int C) {
  int e = blockIdx.x;
  int c = threadIdx.x;  // blockDim.x == H*C == 256
  int s, d;
  edge_sd(e, E, src, dst, s, d);
  int hh = c / C;
  float w = alpha[(size_t)e * H + hh] / (denom[d * H + hh] + 1e-16f);
  atomicAdd(&agg[(size_t)d * (H * C) + c], h[(size_t)s * (H * C) + c] * w);
}

// BN sums: sums[0:D] = sum over nodes, sums[D:2D] = sum of squares
__global__ void bn_stats(const float* __restrict__ x,
                         const float* __restrict__ bias,
                         float* __restrict__ sums, int Nn, int D) {
  int c = threadIdx.x;  // blockDim.x == D
  float s = 0.f, s2 = 0.f;
  for (int n = blockIdx.x; n < Nn; n += gridDim.x) {
    float v = x[(size_t)n * D + c] + bias[c];
    s  += v;
    s2 += v * v;
  }
  atomicAdd(&sums[c], s);
  atomicAdd(&sums[D + c], s2);
}

__global__ void bn_apply(const float* __restrict__ x,
                         const float* __restrict__ bias,
                         const float* __restrict__ sums,
                         const float* __restrict__ gamma,
                         const float* __restrict__ beta,
                         float* __restrict__ y, int Nn, int D, int relu) {
  int t = blockIdx.x * blockDim.x + threadIdx.x;
  if (t >= Nn * D) return;
  int c = t % D;
  float invN = 1.0f / (float)Nn;
  float m = sums[c] * invN;
  float v = sums[D + c] * invN - m * m;
  float val = (x[t] + bias[c] - m) * rsqrtf(v + BN_EPS) * gamma[c] + beta[c];
  if (relu) val = fmaxf(val, 0.f);
  y[t] = val;
}

// ---------------------------------------------------------------------------
static void run_layer(const float* X, int K,
                      const float* W, const float* att_s, const float* att_d,
                      const float* bias, const float* gamma, const float* beta,
                      int H, int C, int relu,
                      float* hbuf, float* aggbuf, float* ybuf,
                      float* a_src, float* a_dst, float* amax, float* denom,
                      float* alpha, float* sums,
                      const long long* esrc, const long long* edst,
                      int E, int Etot, int Nn, hipStream_t stream) {
  int D = H * C;  // 256 in all layers
  // 1) GEMM h = X @ W
  int waves   = (Nn / 16) * (D / 16);
  int gblocks = (waves * 32 + 255) / 256;
  gemm_wmma_f32<<<gblocks, 256, 0, stream>>>(X, W, hbuf, Nn, K, D);
  // 2) attention coefficients
  attn_coef<<<(Nn * H + 255) / 256, 256, 0, stream>>>(hbuf, att_s, att_d,
                                                      a_src, a_dst, Nn, H, C);
  // 3) init reductions
  fill_f32<<<(Nn * H + 255) / 256, 256, 0, stream>>>(amax, -INFINITY, Nn * H);
  fill_f32<<<(Nn * H + 255) / 256, 256, 0, stream>>>(denom, 0.f, Nn * H);
  fill_f32<<<(Nn * D + 255) / 256, 256, 0, stream>>>(aggbuf, 0.f, Nn * D);
  fill_f32<<<(2 * D + 255) / 256, 256, 0, stream>>>(sums, 0.f, 2 * D);
  // 4-5) softmax over incoming edges
  int eth = Etot * H;
  edge_pass1<<<(eth + 255) / 256, 256, 0, stream>>>(esrc, edst, E, Etot,
                                                    a_src, a_dst, alpha, amax, H);
  edge_pass2<<<(eth + 255) / 256, 256, 0, stream>>>(esrc, edst, E, Etot,
                                                    alpha, amax, denom, H);
  // 6) weighted scatter-aggregate
  edge_scatter<<<Etot, D, 0, stream>>>(esrc, edst, E, hbuf, alpha, denom,
                                       aggbuf, H, C);
  // 7-8) bias + batchnorm (+ relu)
  bn_stats<<<256, D, 0, stream>>>(aggbuf, bias, sums, Nn, D);
  bn_apply<<<(Nn * D + 255) / 256, 256, 0, stream>>>(aggbuf, bias, sums,
                                                     gamma, beta, ybuf,
                                                     Nn, D, relu);
}

extern "C" void kernel_launch(void* const* d_in, const int* in_sizes, int n_in,
                              void* d_out, int out_size, void* d_ws, size_t ws_size,
                              hipStream_t stream) {
  (void)n_in; (void)out_size; (void)ws_size;
  const float*     x  = (const float*)d_in[0];
  const long long* ei = (const long long*)d_in[1];  // int64 [2, E]
  const float* W0  = (const float*)d_in[2];
  const float* as0 = (const float*)d_in[3];
  const float* ad0 = (const float*)d_in[4];
  const float* b0  = (const float*)d_in[5];
  const float* g0  = (const float*)d_in[6];
  const float* be0 = (const float*)d_in[7];
  const float* W1  = (const float*)d_in[8];
  const float* as1 = (const float*)d_in[9];
  const float* ad1 = (const float*)d_in[10];
  const float* b1  = (const float*)d_in[11];
  const float* g1  = (const float*)d_in[12];
  const float* be1 = (const float*)d_in[13];
  const float* W2  = (const float*)d_in[14];
  const float* as2 = (const float*)d_in[15];
  const float* ad2 = (const float*)d_in[16];
  const float* b2  = (const float*)d_in[17];
  const float* g2  = (const float*)d_in[18];
  const float* be2 = (const float*)d_in[19];

  const int IN  = 128;
  const int HID = 256;
  int N = in_sizes[0] / IN;   // 10000
  int E = in_sizes[1] / 2;    // 320000
  int Etot = E + N;           // with self-loops

  const long long* esrc = ei;
  const long long* edst = ei + E;

  // workspace layout (floats)
  float* ws    = (float*)d_ws;
  float* P     = ws;                      // N*HID
  float* Q     = P + (size_t)N * HID;     // N*HID
  float* a_src = Q + (size_t)N * HID;     // N*4
  float* a_dst = a_src + (size_t)N * 4;   // N*4
  float* amax  = a_dst + (size_t)N * 4;   // N*4
  float* denom = amax + (size_t)N * 4;    // N*4
  float* alpha = denom + (size_t)N * 4;   // Etot*4
  float* sums  = alpha + (size_t)Etot * 4;// 2*HID

  // Layer 0: x[N,128] -> P (h=P, agg=Q, y=P, relu)
  run_layer(x, IN, W0, as0, ad0, b0, g0, be0, /*H=*/4, /*C=*/64, /*relu=*/1,
            P, Q, P, a_src, a_dst, amax, denom, alpha, sums,
            esrc, edst, E, Etot, N, stream);
  // Layer 1: P[N,256] -> Q (h=Q, agg=P, y=Q, relu)
  run_layer(P, HID, W1, as1, ad1, b1, g1, be1, 4, 64, 1,
            Q, P, Q, a_src, a_dst, amax, denom, alpha, sums,
            esrc, edst, E, Etot, N, stream);
  // Layer 2: Q[N,256] -> d_out (h=P, agg=Q, y=d_out, no relu, heads=1)
  run_layer(Q, HID, W2, as2, ad2, b2, g2, be2, /*H=*/1, /*C=*/256, /*relu=*/0,
            P, Q, (float*)d_out, a_src, a_dst, amax, denom, alpha, sums,
            esrc, edst, E, Etot, N, stream);
}